// FastTextTrainer_7215545057602
// MI455X (gfx1250) — compile-verified
//
#include <hip/hip_runtime.h>

// FastText embedding-bag forward for MI455X (gfx1250, wave32).
// out[b, :] = W_in[center_ids[b], :] + sum_{j in segment b} W_sub[ngram_idx[j], :]
//
// Pure HBM-gather workload: ~433 MB moved, ~103 MFLOP (0.24 FLOP/B) ->
// ~18.6 us floor at 23.3 TB/s. WMMA is inapplicable (no dense matmul).
// CDNA5-specific paths used: wave32 sizing, scalar s_load segment metadata,
// global_load_b128 coalesced row gathers, global_prefetch_b8 lookahead,
// non-temporal (TH=NT) output stores to protect L2 residency of W_sub.

namespace {
constexpr int kD   = 300;       // embedding dim
constexpr int kVec = kD / 4;    // 75 float4 chunks per row (exact)
constexpr int kTPB = 96;        // 3 wave32 per block; lanes 75..95 idle on vector work
}

typedef float ntf4 __attribute__((ext_vector_type(4)));

__device__ __forceinline__ void add4(float4& a, const float4 v) {
    a.x += v.x; a.y += v.y; a.z += v.z; a.w += v.w;
}

// ---------------------------------------------------------------------------
// Prologue: row_start[b] = lower_bound(segment_ids, b), b in [0, nrows].
// One thread per boundary; parallel vector-load binary search (19 steps over
// a 1.3 MB array resident in L2). Replaces ~38 serialized s_load latencies
// per block in the main kernel with one uniform s_load pair.
// ---------------------------------------------------------------------------
__global__ __launch_bounds__(256)
void seg_offsets_kernel(const int* __restrict__ segment_ids, int total,
                        int nrows, int* __restrict__ row_start)
{
    const int b = (int)(blockIdx.x * blockDim.x + threadIdx.x);
    if (b > nrows) return;
    int lo = 0, hi = total;
    while (lo < hi) {
        int m = (lo + hi) >> 1;
        if (segment_ids[m] < b) lo = m + 1; else hi = m;
    }
    row_start[b] = lo;
}

// ---------------------------------------------------------------------------
// Main kernel: one block per output row.
// ---------------------------------------------------------------------------
__global__ __launch_bounds__(kTPB)
void fasttext_embed_bag(const int*   __restrict__ center_ids,
                        const int*   __restrict__ ngram_idx,
                        const int*   __restrict__ segment_ids,
                        const float* __restrict__ W_in,
                        const float* __restrict__ W_sub,
                        float*       __restrict__ out,
                        int total,
                        const int*   __restrict__ row_start)  // nullptr -> inline search
{
    const int b   = (int)blockIdx.x;
    const int tid = (int)threadIdx.x;

    int seg_begin, seg_end;
    if (row_start != nullptr) {
        // Uniform indices -> scalar loads (s_load) of precomputed boundaries.
        seg_begin = row_start[b];
        seg_end   = row_start[b + 1];
    } else {
        // Fallback: uniform binary searches (SALU + s_load path).
        int lo = 0, hi = total;
        while (lo < hi) {
            int m = (lo + hi) >> 1;
            if (segment_ids[m] < b) lo = m + 1; else hi = m;
        }
        seg_begin = lo;
        hi = total;
        while (lo < hi) {
            int m = (lo + hi) >> 1;
            if (segment_ids[m] <= b) lo = m + 1; else hi = m;
        }
        seg_end = lo;
    }

    if (tid >= kVec) return;   // 75 active lanes carry the 300-float row

    // Word vector gather seeds accumulator 0; its load overlaps the loop head.
    const float4* __restrict__ wrow =
        reinterpret_cast<const float4*>(W_in + (size_t)center_ids[b] * kD);
    float4 a0 = wrow[tid];
    float4 a1 = make_float4(0.f, 0.f, 0.f, 0.f);
    float4 a2 = make_float4(0.f, 0.f, 0.f, 0.f);
    float4 a3 = make_float4(0.f, 0.f, 0.f, 0.f);

    int j = seg_begin;

    // 4-way unrolled gather-accumulate: 4 independent b128 gathers in flight,
    // plus global_prefetch_b8 for the entire next unrolled iteration's rows.
    for (; j + 4 <= seg_end; j += 4) {
        const int i0 = ngram_idx[j + 0];
        const int i1 = ngram_idx[j + 1];
        const int i2 = ngram_idx[j + 2];
        const int i3 = ngram_idx[j + 3];

        if (j + 8 <= seg_end) {
            // Each active lane prefetches its own 16B chunk of the 4 rows the
            // next iteration will gather; together the 75 lanes cover all of
            // each row's cachelines. No counters, no data return.
            __builtin_prefetch(W_sub + (size_t)ngram_idx[j + 4] * kD + tid * 4, 0, 3);
            __builtin_prefetch(W_sub + (size_t)ngram_idx[j + 5] * kD + tid * 4, 0, 3);
            __builtin_prefetch(W_sub + (size_t)ngram_idx[j + 6] * kD + tid * 4, 0, 3);
            __builtin_prefetch(W_sub + (size_t)ngram_idx[j + 7] * kD + tid * 4, 0, 3);
        }

        const float4 v0 = reinterpret_cast<const float4*>(W_sub + (size_t)i0 * kD)[tid];
        const float4 v1 = reinterpret_cast<const float4*>(W_sub + (size_t)i1 * kD)[tid];
        const float4 v2 = reinterpret_cast<const float4*>(W_sub + (size_t)i2 * kD)[tid];
        const float4 v3 = reinterpret_cast<const float4*>(W_sub + (size_t)i3 * kD)[tid];

        add4(a0, v0);
        add4(a1, v1);
        add4(a2, v2);
        add4(a3, v3);
    }

    // Tail (segment length averages ~20, not a multiple of 4).
    for (; j < seg_end; ++j) {
        const int i0 = ngram_idx[j];
        const float4 v0 = reinterpret_cast<const float4*>(W_sub + (size_t)i0 * kD)[tid];
        add4(a0, v0);
    }

    add4(a0, a1);
    add4(a2, a3);
    add4(a0, a2);

    // Output is written once and never re-read: non-temporal b128 store keeps
    // W_sub rows resident in the 192 MB L2 instead.
    ntf4 r;
    r.x = a0.x; r.y = a0.y; r.z = a0.z; r.w = a0.w;
    __builtin_nontemporal_store(r, reinterpret_cast<ntf4*>(out + (size_t)b * kD) + tid);
}

extern "C" void kernel_launch(void* const* d_in, const int* in_sizes, int n_in,
                              void* d_out, int out_size, void* d_ws, size_t ws_size,
                              hipStream_t stream) {
    // setup_inputs() order:
    //   0: center_ids  (B)        int32
    //   1: ngram_idx   (TOTAL)    int32
    //   2: segment_ids (TOTAL)    int32, sorted
    //   3: W_in_weight (VOCAB*D)  float32
    //   4: W_sub_weight(BUCKET*D) float32
    const int*   center = (const int*)  d_in[0];
    const int*   ngram  = (const int*)  d_in[1];
    const int*   seg    = (const int*)  d_in[2];
    const float* w_in   = (const float*)d_in[3];
    const float* w_sub  = (const float*)d_in[4];
    float*       out    = (float*)      d_out;

    const int B     = in_sizes[0];   // 16384 output rows
    const int total = in_sizes[1];   // 327680 ngram entries

    int* row_start = nullptr;
    if (ws_size >= (size_t)(B + 1) * sizeof(int)) {
        row_start = (int*)d_ws;
        const int nb = (B + 1 + 255) / 256;
        seg_offsets_kernel<<<dim3((unsigned)nb), dim3(256), 0, stream>>>(
            seg, total, B, row_start);
    }

    fasttext_embed_bag<<<dim3((unsigned)B), dim3(kTPB), 0, stream>>>(
        center, ngram, seg, w_in, w_sub, out, total, row_start);
}